// GraphResidualBlock_59511066853466
// MI455X (gfx1250) — compile-verified
//
#include <hip/hip_runtime.h>

// GraphResidualBlock for MI455X (gfx1250).
// out[b,m] = x[b,m] + b2 + f(s[b,m]),  s = softmax(A_param) @ x[b,:]
// f(s) = sum_h W2[h]*relu(W1[h]*s + b1[h])  -- exact piecewise-linear, 33 segments.
// Mixing (s = A_sm @ x) done with V_WMMA_F32_16X16X4_F32 (full f32 precision).

#define Bn 1048576
#define Nn 6
#define Hh 32

typedef __attribute__((ext_vector_type(2))) float v2f;
typedef __attribute__((ext_vector_type(8))) float v8f;

__device__ __forceinline__ v8f wmma4(v2f a, v2f b, v8f c) {
    // 8 args: (neg_a, A, neg_b, B, c_mod, C, reuse_a, reuse_b)
    return __builtin_amdgcn_wmma_f32_16x16x4_f32(false, a, false, b, (short)0, c,
                                                 false, false);
}

// ---------------------------------------------------------------------------
// ws layout (float units):
//   [0,32)    sorted knots
//   [32,98)   33 x (slope, intercept) pairs  (byte offset 128, 8-aligned)
//   [112,240) per-lane WMMA B-operand float4  (byte offset 448, 16-aligned)
// ---------------------------------------------------------------------------

__global__ void precompute_kernel(const float* __restrict__ A_param,
                                  const float* __restrict__ W1,
                                  const float* __restrict__ b1,
                                  const float* __restrict__ W2,
                                  float* __restrict__ ws) {
    __shared__ float As[6][6];   // softmax(A_param) rows
    __shared__ float kn[32];
    __shared__ float skn[32];
    int t = threadIdx.x;

    if (t < 6) {
        float r[6]; float mx = -3.4e38f;
        for (int j = 0; j < 6; ++j) { r[j] = A_param[t * 6 + j]; mx = fmaxf(mx, r[j]); }
        float sum = 0.f;
        for (int j = 0; j < 6; ++j) { r[j] = __expf(r[j] - mx); sum += r[j]; }
        float inv = 1.f / sum;
        for (int j = 0; j < 6; ++j) As[t][j] = r[j] * inv;
    }
    if (t < 32) {
        float w = W1[t], b = b1[t];
        kn[t] = (w != 0.f) ? (-b / w) : 3.0e38f;   // W1==0: knot never crossed
    }
    __syncthreads();
    if (t < 32) {
        // rank sort (stable), O(32^2) total
        float k = kn[t]; int rank = 0;
        for (int j = 0; j < 32; ++j) {
            float kj = kn[j];
            rank += (kj < k) || (kj == k && j < t);
        }
        skn[rank] = k;
    }
    __syncthreads();
    if (t < 32) ws[t] = skn[t];
    if (t < 33) {
        float rep;
        if (t == 0)       rep = skn[0] - 1.f;
        else if (t == 32) rep = skn[31] + 1.f;
        else              rep = skn[t - 1] + 0.5f * (skn[t] - skn[t - 1]);
        float sl = 0.f, ic = 0.f;
        for (int h = 0; h < 32; ++h) {
            float w = W1[h], b = b1[h];
            if (w * rep + b > 0.f) { sl += W2[h] * w; ic += W2[h] * b; }
        }
        ws[32 + 2 * t] = sl;
        ws[33 + 2 * t] = ic;
    }
    if (t < 32) {
        // B-operand for the mixing WMMAs, per lane.
        // B 4x16 layout assumption: vgpr j, half hi -> row K = j + 2*hi, col = lane%16.
        // Bm[k][n] = As[n][k] (n = output node), zero-padded.
        int lo = t & 15, hi = t >> 4;
        float v0 = (lo < 6 && (0 + 2 * hi) < 6) ? As[lo][0 + 2 * hi] : 0.f;
        float v1 = (lo < 6 && (1 + 2 * hi) < 6) ? As[lo][1 + 2 * hi] : 0.f;
        float v2 = (lo < 6 && (4 + 2 * hi) < 6) ? As[lo][4 + 2 * hi] : 0.f;
        float v3 = (lo < 6 && (5 + 2 * hi) < 6) ? As[lo][5 + 2 * hi] : 0.f;
        float4 v; v.x = v0; v.y = v1; v.z = v2; v.w = v3;
        ((float4*)(ws + 112))[t] = v;
    }
}

// ---------------------------------------------------------------------------
// Main kernel: one wave handles a 32-sample tile per iteration.
// ---------------------------------------------------------------------------
__global__ __launch_bounds__(256) void graph_main(const float* __restrict__ x,
                                                  const float* __restrict__ b2,
                                                  const float* __restrict__ ws,
                                                  float* __restrict__ out) {
    __shared__ float            sK[32];
    __shared__ float2           sSeg[33];
    __shared__ __align__(16) float sX[8][192];  // per-wave x tile  [32][6]
    __shared__ __align__(16) float sS[8][192];  // per-wave s tile  [32][6]

    int tid = threadIdx.x;
    if (tid < 32) sK[tid] = ws[tid];
    if (tid < 33) sSeg[tid] = make_float2(ws[32 + 2 * tid], ws[33 + 2 * tid]);
    __syncthreads();

    int l  = tid & 31;
    int wv = tid >> 5;
    int lo = l & 15;
    int hi = l >> 4;

    float4 bm = ((const float4*)(ws + 112))[l];
    v2f bc0 = {bm.x, bm.y};   // mixing chunk K=0..3
    v2f bc1 = {bm.z, bm.w};   // mixing chunk K=4..5 (+zero pad)
    float b2v = b2[0];

    float* X = sX[wv];
    float* S = sS[wv];

    int nwaves = (gridDim.x * blockDim.x) >> 5;
    int wid    = (blockIdx.x * blockDim.x + threadIdx.x) >> 5;

    for (int T = wid; T < Bn / 32; T += nwaves) {
        // ---- coalesced global -> LDS copy of 32 samples (768 B) ----
        const float* xg = x + (size_t)T * 192;
        ((float4*)X)[l] = ((const float4*)xg)[l];
        if (l < 16) ((float4*)X)[32 + l] = ((const float4*)xg)[32 + l];

        // ---- node mixing via f32 WMMA (two 16-sample tiles) ----
        const float2* X2 = (const float2*)X;
        const v2f zero2 = {0.f, 0.f};
#pragma unroll
        for (int t2 = 0; t2 < 2; ++t2) {
            // A 16x4 layout: lane lo = row M; vgpr0 = K(0|2), vgpr1 = K(1|3)
            float2 a0f = X2[(t2 * 16 + lo) * 3 + hi]; // (x0,x1) lo-half, (x2,x3) hi-half
            float2 a1f = X2[(t2 * 16 + lo) * 3 + 2];  // (x4,x5); zero in hi-half
            v2f a0 = {a0f.x, a0f.y};
            v2f a1 = hi ? zero2 : (v2f){a1f.x, a1f.y};
            v8f acc = {0.f, 0.f, 0.f, 0.f, 0.f, 0.f, 0.f, 0.f};
            acc = wmma4(a1, bc1, acc);
            acc = wmma4(a0, bc0, acc);
            // D layout: vgpr v, half hi -> row m = v + 8*hi, col n = lo
            if (lo < 6) {
#pragma unroll
                for (int v = 0; v < 8; ++v)
                    S[(t2 * 16 + v + 8 * hi) * 6 + lo] = acc[v];
            }
        }

        // ---- per-lane piecewise-linear f(s) + residual (sample = lane) ----
        float xv[6], sv[6];
        const float2* S2 = (const float2*)S;
#pragma unroll
        for (int j = 0; j < 3; ++j) {
            float2 p = S2[l * 3 + j]; sv[2 * j] = p.x; sv[2 * j + 1] = p.y;
            float2 q = X2[l * 3 + j]; xv[2 * j] = q.x; xv[2 * j + 1] = q.y;
        }
        float2 o[3];
#pragma unroll
        for (int n = 0; n < 6; ++n) {
            float s = sv[n];
            int pos = 0;
#pragma unroll
            for (int st = 16; st >= 1; st >>= 1) {
                if (sK[pos + st - 1] < s) pos += st;   // pos = #knots < s
            }
            float2 sg = sSeg[pos];
            float f = fmaf(sg.x, s, sg.y);
            float r = xv[n] + b2v + f;
            if (n == 0) o[0].x = r; else if (n == 1) o[0].y = r;
            else if (n == 2) o[1].x = r; else if (n == 3) o[1].y = r;
            else if (n == 4) o[2].x = r; else o[2].y = r;
        }
        float2* og = (float2*)(out + (size_t)T * 192);
        og[l * 3 + 0] = o[0];
        og[l * 3 + 1] = o[1];
        og[l * 3 + 2] = o[2];
    }
}

extern "C" void kernel_launch(void* const* d_in, const int* in_sizes, int n_in,
                              void* d_out, int out_size, void* d_ws, size_t ws_size,
                              hipStream_t stream) {
    (void)in_sizes; (void)n_in; (void)out_size; (void)ws_size;
    const float* x       = (const float*)d_in[0];
    const float* A_param = (const float*)d_in[1];
    const float* W1      = (const float*)d_in[2];
    const float* b1      = (const float*)d_in[3];
    const float* W2      = (const float*)d_in[4];
    const float* b2      = (const float*)d_in[5];
    float* out = (float*)d_out;
    float* ws  = (float*)d_ws;

    precompute_kernel<<<1, 64, 0, stream>>>(A_param, W1, b1, W2, ws);
    graph_main<<<1024, 256, 0, stream>>>(x, b2, ws, out);
}